// AttentionSubgraphExtractor_55379308315327
// MI455X (gfx1250) — compile-verified
//
#include <hip/hip_runtime.h>
#include <hip/hip_bf16.h>

#define NUM_MHC 300
#define NUM_PEP 24
#define SIZE_N  324
#define TOPK    20
#define DD      512

typedef float v2f __attribute__((ext_vector_type(2)));
typedef float v4f __attribute__((ext_vector_type(4)));
typedef float v8f __attribute__((ext_vector_type(8)));

// 300-bit membership bitmap for _INDICES (34 entries)
__device__ __constant__ unsigned kIdxBitmap[10] = {
    0x01000280u, 0xC8002000u, 0x8013366Cu, 0x0054000Au, 0xD1488000u,
    0x00000888u, 0x00000000u, 0x00000000u, 0x00000000u, 0x00000000u};

__device__ __forceinline__ bool is_idx(int m) {
  return (kIdxBitmap[m >> 5] >> (m & 31)) & 1u;
}

// ---------------------------------------------------------------------------
// Kernel 0: w_eff[d] = sum_a Wk[a,d] * a2[a]   (512 outputs, trivial)
// ---------------------------------------------------------------------------
__global__ __launch_bounds__(256) void weff_kernel(const float* __restrict__ Wk,
                                                   const float* __restrict__ av,
                                                   float* __restrict__ weff) {
  int d = blockIdx.x * 256 + threadIdx.x;
  if (d < DD) {
    float acc = 0.f;
#pragma unroll 8
    for (int a = 0; a < 64; ++a) acc = fmaf(Wk[a * DD + d], av[64 + a], acc);
    weff[d] = acc;
  }
}

// ---------------------------------------------------------------------------
// Kernel 1: per-batch scores s[p] = H_p[b,p,:]·w_eff via V_WMMA_F32_16X16X4_F32
// (B fragment columns replicated with w => C[m,n] = dot(H_p row m, w)),
// then rank-based top-20 set -> 24-bit mask per batch.
// One wave32 per batch; EXEC all-ones through the WMMA loop.
// ---------------------------------------------------------------------------
__global__ __launch_bounds__(32) void score_topk_kernel(
    const float* __restrict__ Hp, const float* __restrict__ weff,
    unsigned* __restrict__ maskOut) {
  const int b    = blockIdx.x;
  const int lane = threadIdx.x;
  const float* hp = Hp + (size_t)b * NUM_PEP * DD;

  const int half = lane >> 4;       // 0: K0/K1 ; 1: K2/K3
  const int m    = lane & 15;
  const int koff = half * 2;
  const int row0 = m;                                      // tile0: rows 0..15
  const int row1 = (16 + m < NUM_PEP) ? (16 + m) : 0;      // tile1: rows 16..23 (pad dup)

  v8f acc0 = {};
  v8f acc1 = {};
  for (int k = 0; k < DD; k += 4) {
    const int kk = k + koff;
    v2f bf;  bf.x = weff[kk];             bf.y = weff[kk + 1];
    v2f a0;  a0.x = hp[row0 * DD + kk];   a0.y = hp[row0 * DD + kk + 1];
    v2f a1;  a1.x = hp[row1 * DD + kk];   a1.y = hp[row1 * DD + kk + 1];
    acc0 = __builtin_amdgcn_wmma_f32_16x16x4_f32(false, a0, false, bf,
                                                 (short)0, acc0, false, false);
    acc1 = __builtin_amdgcn_wmma_f32_16x16x4_f32(false, a1, false, bf,
                                                 (short)0, acc1, false, false);
  }

  // C layout: VGPR j holds row M=j (lanes 0-15) and M=j+8 (lanes 16-31); all
  // columns equal, so lane 0 / lane 16 each own 8 row-results per tile.
  __shared__ float s[32];
  if (lane == 0) {
#pragma unroll
    for (int j = 0; j < 8; ++j) { s[j] = acc0[j]; s[16 + j] = acc1[j]; }
  }
  if (lane == 16) {
#pragma unroll
    for (int j = 0; j < 8; ++j) { s[8 + j] = acc0[j]; }
  }
  __syncthreads();

  // rank with lowest-index tie-break == jax.lax.top_k semantics; keep rank<20
  unsigned flag = 0;
  if (lane < NUM_PEP) {
    const float v = s[lane];
    int cnt = 0;
#pragma unroll
    for (int q = 0; q < NUM_PEP; ++q) {
      const float u = s[q];
      cnt += (u > v) || (u == v && q < lane);
    }
    flag = (cnt < TOPK) ? 1u : 0u;
  }
  unsigned long long bal = __ballot(flag);
  if (lane == 0) maskOut[b] = (unsigned)bal;
}

// ---------------------------------------------------------------------------
// Kernel 2: A_sub fill (215 MB), values computed on the fly, NT b128 stores.
//   row<300, col<300 : identity
//   row<300, col>=300: t=col-300; is_idx(row) ? bit(t) : (t<20)
//   row>=300,col<300 : t=row-300; is_idx(col) ? bit(t) : (t<20)
//   else 0
// ---------------------------------------------------------------------------
__global__ __launch_bounds__(256) void build_asub_kernel(
    const unsigned* __restrict__ masks, float* __restrict__ outA) {
  const int PLANE4 = (SIZE_N * SIZE_N) / 4;  // 26244 float4 per batch
  const int b = blockIdx.y;
  const int i = blockIdx.x * 256 + threadIdx.x;
  if (i >= PLANE4) return;
  const unsigned mk = masks[b];
  const int row = i / (SIZE_N / 4);          // 81 float4 per row
  const int c0  = (i % (SIZE_N / 4)) * 4;

  v4f v;
  if (row < NUM_MHC) {
    const bool ridx = is_idx(row);
#pragma unroll
    for (int j = 0; j < 4; ++j) {
      const int c = c0 + j;
      float x;
      if (c < NUM_MHC) {
        x = (c == row) ? 1.f : 0.f;
      } else {
        const int t = c - NUM_MHC;
        x = (ridx ? ((mk >> t) & 1u) != 0u : (t < TOPK)) ? 1.f : 0.f;
      }
      v[j] = x;
    }
  } else {
    const int  t    = row - NUM_MHC;
    const bool trow = ((mk >> t) & 1u) != 0u;
#pragma unroll
    for (int j = 0; j < 4; ++j) {
      const int c = c0 + j;
      float x = 0.f;
      if (c < NUM_MHC) x = (is_idx(c) ? trow : (t < TOPK)) ? 1.f : 0.f;
      v[j] = x;
    }
  }
  float* dst = outA + (size_t)b * (SIZE_N * SIZE_N) + (size_t)row * SIZE_N + c0;
  __builtin_nontemporal_store(v, (v4f*)dst);
}

// ---------------------------------------------------------------------------
// Kernel 3: H_sub = concat(H_m, H_p) per batch, NT b128 streaming copy.
// 41472 float4 per batch; grid.x * 256 covers it exactly.
// ---------------------------------------------------------------------------
__global__ __launch_bounds__(256) void build_hsub_kernel(
    const float* __restrict__ Hm, const float* __restrict__ Hp,
    float* __restrict__ outH) {
  const int b   = blockIdx.y;
  const int i   = blockIdx.x * 256 + threadIdx.x;  // float4 index in plane
  const int row = i >> 7;                           // 128 float4 per 512-row
  const int c0  = (i & 127) * 4;
  const float* src = (row < NUM_MHC)
      ? (Hm + (size_t)b * NUM_MHC * DD + (size_t)row * DD + c0)
      : (Hp + (size_t)b * NUM_PEP * DD + (size_t)(row - NUM_MHC) * DD + c0);
  v4f v = __builtin_nontemporal_load((const v4f*)src);
  __builtin_nontemporal_store(
      v, (v4f*)(outH + (size_t)b * SIZE_N * DD + (size_t)row * DD + c0));
}

// ---------------------------------------------------------------------------
extern "C" void kernel_launch(void* const* d_in, const int* in_sizes, int n_in,
                              void* d_out, int out_size, void* d_ws,
                              size_t ws_size, hipStream_t stream) {
  // setup_inputs order: H_m, H_p, A_mm, Wq, bq, Wk, bk, att_vec
  const float* Hm = (const float*)d_in[0];
  const float* Hp = (const float*)d_in[1];
  const float* Wk = (const float*)d_in[5];
  const float* av = (const float*)d_in[7];
  const int B = in_sizes[0] / (NUM_MHC * DD);

  float*    outA = (float*)d_out;                                 // B*324*324
  float*    outH = outA + (size_t)B * SIZE_N * SIZE_N;            // B*324*512
  float*    weff  = (float*)d_ws;                                 // 512 floats
  unsigned* masks = (unsigned*)((char*)d_ws + DD * sizeof(float)); // B uint32

  weff_kernel<<<dim3((DD + 255) / 256), 256, 0, stream>>>(Wk, av, weff);
  score_topk_kernel<<<dim3(B), 32, 0, stream>>>(Hp, weff, masks);

  const int plane4A = (SIZE_N * SIZE_N) / 4;  // 26244
  build_asub_kernel<<<dim3((plane4A + 255) / 256, B), 256, 0, stream>>>(masks,
                                                                        outA);
  const int plane4H = (SIZE_N * DD) / 4;      // 41472 == 162*256
  build_hsub_kernel<<<dim3(plane4H / 256, B), 256, 0, stream>>>(Hm, Hp, outH);
}